// TEGConv_88227218195146
// MI455X (gfx1250) — compile-verified
//
#include <hip/hip_runtime.h>

typedef __attribute__((ext_vector_type(2))) float v2f;
typedef __attribute__((ext_vector_type(8))) float v8f;

#define N_NODES 50000
#define N_EDGES 800000
#define XDIM    64
#define EDIM    32
#define KDIM    96
#define ODIM    64
#define KSTEPS  24   // KDIM / 4
#define WELEMS  (ODIM * KDIM)   // 6144 floats = 24 KB

// ---------------------------------------------------------------------------
// Kernel 0: zero the accumulation buffers (d_out sums + counts in workspace)
// ---------------------------------------------------------------------------
__global__ __launch_bounds__(256) void tegconv_zero(float* __restrict__ out,
                                                    float* __restrict__ cnt) {
    const long long total = (long long)N_NODES * ODIM + N_NODES;
    long long i = (long long)blockIdx.x * blockDim.x + threadIdx.x;
    if (i < (long long)N_NODES * ODIM) {
        out[i] = 0.0f;
    } else if (i < total) {
        cnt[i - (long long)N_NODES * ODIM] = 0.0f;
    }
}

// ---------------------------------------------------------------------------
// Kernel 1: per-edge linear via fp32 WMMA + atomic scatter-add
// One wave32 handles a tile of 16 edges. 6250 blocks x 8 waves = 50000 tiles.
// ---------------------------------------------------------------------------
__global__ __launch_bounds__(256) void tegconv_edges(
    const float* __restrict__ x,       // [N_NODES, 64]
    const int*   __restrict__ src,     // [E]
    const int*   __restrict__ dst,     // [E]
    const float* __restrict__ ef,      // [E, 32]
    const float* __restrict__ W,       // [64, 96] row-major
    const float* __restrict__ bias,    // [64]
    float*       __restrict__ out,     // [N_NODES, 64] (sums, pre-zeroed)
    float*       __restrict__ cnt)     // [N_NODES]     (counts, pre-zeroed)
{
    __shared__ float lw[WELEMS];

    // Stage W into LDS (uniform trip count: 6144 / 256 = 24 per thread).
    for (int i = threadIdx.x; i < WELEMS; i += 256) {
        lw[i] = W[i];
    }
    __syncthreads();

    const int lane = threadIdx.x & 31;
    const int wave = threadIdx.x >> 5;
    const int tile = blockIdx.x * 8 + wave;   // 0..49999
    const int base = tile << 4;               // first edge of the tile
    const int mrow = lane & 15;               // M (edge row) for A / N for B
    const int half = lane >> 4;               // lane half selects K/M offset

    // ---- Preload A fragments for all 24 K-steps -------------------------
    // A-layout (16x4 f32): lane L, vgpr v holds A[M=L%16][K = 4*j + v + 2*half]
    v2f afrag[KSTEPS];
    {
        const int  e  = base + mrow;
        const int  s  = src[e];
        const float* xr = x + (long long)s * XDIM + 2 * half;
        #pragma unroll
        for (int j = 0; j < 16; ++j) {
            afrag[j] = *reinterpret_cast<const v2f*>(xr + j * 4);
        }
        const float* er = ef + (long long)e * EDIM + 2 * half;
        #pragma unroll
        for (int j = 0; j < 8; ++j) {
            afrag[16 + j] = *reinterpret_cast<const v2f*>(er + j * 4);
        }
    }

    // ---- Destination node ids for the 8 C-rows this lane owns -----------
    // C-layout: vgpr v holds C[M = v + 8*half][N = mrow]
    int dnode[8];
    #pragma unroll
    for (int v = 0; v < 8; ++v) {
        dnode[v] = dst[base + v + 8 * half];
    }

    // ---- 4 N-tiles of 16 output columns ---------------------------------
    #pragma unroll
    for (int nt = 0; nt < 4; ++nt) {
        const int n = nt * 16 + mrow;                 // output column
        const float* wr = lw + n * KDIM + 2 * half;   // B[k][n] = W[n][k]

        v8f c = {0.f, 0.f, 0.f, 0.f, 0.f, 0.f, 0.f, 0.f};
        #pragma unroll
        for (int j = 0; j < KSTEPS; ++j) {
            v2f bf = *reinterpret_cast<const v2f*>(wr + j * 4);
            c = __builtin_amdgcn_wmma_f32_16x16x4_f32(
                    /*neg_a=*/false, afrag[j],
                    /*neg_b=*/false, bf,
                    /*c_mod=*/(short)0, c,
                    /*reuse_a=*/false, /*reuse_b=*/false);
        }

        const float bv = bias[n];
        #pragma unroll
        for (int v = 0; v < 8; ++v) {
            atomicAdd(out + (long long)dnode[v] * ODIM + n, c[v] + bv);
        }
    }

    // ---- Edge counts (after all WMMAs; divergence here is safe) ---------
    if (lane < 16) {
        atomicAdd(cnt + dst[base + lane], 1.0f);
    }
}

// ---------------------------------------------------------------------------
// Kernel 2: sums / max(count, 1)
// ---------------------------------------------------------------------------
__global__ __launch_bounds__(256) void tegconv_finalize(float* __restrict__ out,
                                                        const float* __restrict__ cnt) {
    const int i = blockIdx.x * blockDim.x + threadIdx.x;
    if (i < N_NODES * ODIM) {
        const float c = cnt[i >> 6];   // i / 64
        out[i] = out[i] / fmaxf(c, 1.0f);
    }
}

// ---------------------------------------------------------------------------
extern "C" void kernel_launch(void* const* d_in, const int* in_sizes, int n_in,
                              void* d_out, int out_size, void* d_ws, size_t ws_size,
                              hipStream_t stream) {
    const float* x  = (const float*)d_in[0];           // [50000, 64]
    const int*   ei = (const int*)  d_in[1];           // [2, 800000] int32
    const float* ef = (const float*)d_in[2];           // [800000, 32]
    const float* W  = (const float*)d_in[3];           // [64, 96]
    const float* b  = (const float*)d_in[4];           // [64]

    float* out = (float*)d_out;                        // [50000, 64]
    float* cnt = (float*)d_ws;                         // [50000] scratch

    const int* src = ei;
    const int* dst = ei + N_EDGES;

    // Zero sums + counts (3,250,000 elements)
    {
        const long long total = (long long)N_NODES * ODIM + N_NODES;
        const int blocks = (int)((total + 255) / 256);
        tegconv_zero<<<blocks, 256, 0, stream>>>(out, cnt);
    }

    // Edge kernel: 50000 tiles of 16 edges, 8 waves per block
    tegconv_edges<<<N_EDGES / (16 * 8), 256, 0, stream>>>(
        x, src, dst, ef, W, b, out, cnt);

    // Finalize: divide by counts
    {
        const int total = N_NODES * ODIM;
        tegconv_finalize<<<(total + 255) / 256, 256, 0, stream>>>(out, cnt);
    }
}